// TimeMix_6_87505663689458
// MI455X (gfx1250) — compile-verified
//
#include <hip/hip_runtime.h>

// ============================================================================
// RWKV-6 TimeMix forward for MI455X (gfx1250, wave32, WMMA + Tensor Data Mover).
// Heavy GEMMs: V_WMMA_F32_16X16X4_F32 (full f32 fidelity vs f32 reference),
// tiles staged LDS-side by TENSOR_LOAD_TO_LDS (double buffered, TENSORcnt).
// ============================================================================

#define Bq   4
#define Tq   1024
#define Cq   1024
#define Hq   16
#define HD   64
#define DMIX 32
#define DDEC 64
#define BT   (Bq * Tq)               // 4096 tokens
#define EPSGN 6.4e-4f                // 1e-5 * 8^2

typedef __attribute__((ext_vector_type(2))) float v2f;
typedef __attribute__((ext_vector_type(8))) float v8f;
typedef unsigned int u32x4 __attribute__((ext_vector_type(4)));
typedef int          i32x4 __attribute__((ext_vector_type(4)));
typedef int          i32x8 __attribute__((ext_vector_type(8)));

#if __has_builtin(__builtin_amdgcn_tensor_load_to_lds) && \
    __has_builtin(__builtin_amdgcn_s_wait_tensorcnt)
#define HAVE_TDM 1
#else
#define HAVE_TDM 0
#endif

// ----------------------------------------------------------------------------
// 1) token shift:  xx = shift(x) - x ;  z = x + xx * time_maa_x
// ----------------------------------------------------------------------------
__global__ __launch_bounds__(256)
void prep_shift_mix(const float* __restrict__ x, const float* __restrict__ tmx,
                    float* __restrict__ xx, float* __restrict__ z)
{
    long i = (long)blockIdx.x * blockDim.x + threadIdx.x;
    if (i >= (long)BT * Cq) return;
    int  c = (int)(i & (Cq - 1));
    long t = (i >> 10) & (Tq - 1);          // Cq == 1024
    float xi = x[i];
    float xp = (t > 0) ? x[i - Cq] : 0.0f;
    float d  = xp - xi;
    xx[i] = d;
    z[i]  = fmaf(d, tmx[c], xi);
}

#if HAVE_TDM
// ----------------------------------------------------------------------------
// Tensor Data Mover: 2D tile (tile_w x tile_h elems, 4B) global -> LDS with
// per-row pad so LDS rows land on the padded (bank-conflict-free) strides.
// D# built per CDNA5 ISA 8.3/8.4:  group0 = {count/lds/global/type},
// group1 = {data_size, pad, dims, strides}.  tensor_w < tile_w => HW zero-fill.
// pad_icode: interval 2^(c+1) DWORDs ; pad_acode: (c+1) DWORDs padding.
// Toolchain: 6-arg builtin (clang-23 form): (g0, g1, g2, g3, g4, cpol).
// ----------------------------------------------------------------------------
__device__ __forceinline__
void tdm_load_2d(unsigned lds_off, const float* gptr,
                 int tile_w, int tile_h, int tensor_w, int stride_elems,
                 int pad_icode, int pad_acode)
{
    unsigned long long ga = (unsigned long long)(uintptr_t)gptr;
    u32x4 g0;
    g0[0] = 1u;                                        // count=1, user desc
    g0[1] = lds_off;                                   // LDS byte address
    g0[2] = (unsigned)ga;                              // global addr lo
    g0[3] = (unsigned)((ga >> 32) & 0x1FFFFFFu) | (2u << 30);  // hi + type=2
    i32x8 g1;
    g1[0] = (2 << 16) | (1 << 20) | (pad_icode << 22) | (pad_acode << 25);
    g1[1] = (tensor_w & 0xFFFF) << 16;                 // tensor_dim0 [63:48]
    g1[2] = ((tensor_w >> 16) & 0xFFFF) | ((tile_h & 0xFFFF) << 16); // td0 hi | tensor_dim1 lo
    g1[3] = ((tile_h >> 16) & 0xFFFF) | ((tile_w & 0xFFFF) << 16);   // td1 hi | tile_dim0
    g1[4] = tile_h & 0xFFFF;                           // tile_dim1 (tile_dim2=0)
    g1[5] = stride_elems;                              // tensor_dim0_stride lo32
    g1[6] = 0;
    g1[7] = 0;
    i32x4 gz4 = {0, 0, 0, 0};                          // 2D: groups 2/3 unused
    i32x8 gz8 = {0, 0, 0, 0, 0, 0, 0, 0};
    __builtin_amdgcn_tensor_load_to_lds(g0, g1, gz4, gz4, gz8, 0);
}
#endif

// ----------------------------------------------------------------------------
// 2) Generic f32 WMMA GEMM:  C[M,N] = epi( A[M,K] * B[K,N] )
//    Workgroup: 256 thr (8 wave32s), 128x64 tile; wave w -> rows [16w,16w+16),
//    4 accumulators covering all 64 cols (A fragment reused 4x).
//    K staged 32-deep; TDM double-buffered (1 barrier/chunk) or manual fallback.
//    Call-site invariants: M % 128 == 0, K % 32 == 0; N guarded (160/64/1024).
//    Epilogues: 0 none | 1 tanh | 2 silu | 3 exp(-exp(bias[n]+acc))
//               4 e0 + e1*(bias[n]+acc)   (fused token-mix output)
// ----------------------------------------------------------------------------
#define GEMM_COMPUTE(cur)                                                     \
    _Pragma("unroll")                                                         \
    for (int kk = 0; kk < 32; kk += 4) {                                      \
        const int kb = kk + (half << 1);                                      \
        v2f a;  a.x = As[cur][wm + l16][kb];  a.y = As[cur][wm + l16][kb + 1];\
        _Pragma("unroll")                                                     \
        for (int t = 0; t < 4; ++t) {                                         \
            v2f b;                                                            \
            b.x = Bs[cur][kb    ][(t << 4) + l16];                            \
            b.y = Bs[cur][kb + 1][(t << 4) + l16];                            \
            acc[t] = __builtin_amdgcn_wmma_f32_16x16x4_f32(                   \
                false, a, false, b, (short)0, acc[t], false, false);          \
        }                                                                     \
    }

__global__ __launch_bounds__(256)
void wmma_gemm_f32(const float* __restrict__ A, int lda,
                   const float* __restrict__ Bm, int ldb,
                   float* __restrict__ Cc, int ldc,
                   int M, int N, int K,
                   int epi, const float* __restrict__ bias,
                   const float* __restrict__ e0, const float* __restrict__ e1)
{
#if HAVE_TDM
    __shared__ float As[2][128][36];   // 128 x 32 rows, +4 DW pad (TDM pad_enable)
    __shared__ float Bs[2][32][68];    // 32 x 64 rows,  +4 DW pad
#else
    __shared__ float As[1][128][36];
    __shared__ float Bs[1][32][68];
#endif
    const int tid  = threadIdx.x;
    const int lane = tid & 31;
    const int wave = tid >> 5;
    const int wm   = wave << 4;          // wave row offset: 0..112
    const int half = lane >> 4;          // lane half selects K pair / M+8
    const int l16  = lane & 15;
    const int m0   = blockIdx.y << 7;    // 128-row tiles
    const int n0   = blockIdx.x << 6;    // 64-col tiles
    const int nw   = (N - n0 < 64) ? (N - n0) : 64;   // valid cols this tile

    v8f acc[4] = { {}, {}, {}, {} };

#if HAVE_TDM
    const unsigned aOff[2] = { (unsigned)(uintptr_t)&As[0][0][0],
                               (unsigned)(uintptr_t)&As[1][0][0] };
    const unsigned bOff[2] = { (unsigned)(uintptr_t)&Bs[0][0][0],
                               (unsigned)(uintptr_t)&Bs[1][0][0] };
    const int nch = K >> 5;
    if (wave == 0) {   // one DMA pair per workgroup per chunk
        tdm_load_2d(aOff[0], A + (long)m0 * lda, 32, 128, 32, lda, 4, 3);
        tdm_load_2d(bOff[0], Bm + n0,            64, 32,  nw, ldb, 5, 3);
    }
    for (int ch = 0; ch < nch; ++ch) {
        const int cur = ch & 1;
        if (wave == 0) __builtin_amdgcn_s_wait_tensorcnt(0);
        __syncthreads();                    // chunk data visible; prev reads done
        if (wave == 0 && ch + 1 < nch) {    // prefetch next chunk into other buf
            const int k1 = (ch + 1) << 5;
            tdm_load_2d(aOff[cur ^ 1], A + (long)m0 * lda + k1, 32, 128, 32, lda, 4, 3);
            tdm_load_2d(bOff[cur ^ 1], Bm + (long)k1 * ldb + n0, 64, 32, nw, ldb, 5, 3);
        }
        GEMM_COMPUTE(cur)
    }
#else
    for (int k0 = 0; k0 < K; k0 += 32) {
        #pragma unroll
        for (int j = 0; j < 16; ++j) {      // A tile 128x32
            int idx = tid + (j << 8);
            int r = idx >> 5, c = idx & 31;
            As[0][r][c] = A[(long)(m0 + r) * lda + (k0 + c)];
        }
        #pragma unroll
        for (int j = 0; j < 8; ++j) {       // B tile 32x64, zero-fill OOB cols
            int idx = tid + (j << 8);
            int r = idx >> 6, c = idx & 63;
            Bs[0][r][c] = (c < nw) ? Bm[(long)(k0 + r) * ldb + (n0 + c)] : 0.0f;
        }
        if (k0 + 32 < K) {
            __builtin_prefetch(&A [(long)(m0 + (tid >> 1)) * lda + (k0 + 32)], 0, 0);
            __builtin_prefetch(&Bm[(long)(k0 + 32 + (tid >> 5)) * ldb + n0  ], 0, 0);
        }
        __syncthreads();
        GEMM_COMPUTE(0)
        __syncthreads();
    }
#endif

    // store with fused epilogue; C/D layout: M = i + 8*half, N = lane&15
    #pragma unroll
    for (int i = 0; i < 8; ++i) {
        const int m = m0 + wm + (half << 3) + i;
        #pragma unroll
        for (int t = 0; t < 4; ++t) {
            const int n = n0 + (t << 4) + l16;
            if (n >= N) continue;
            float a = acc[t][i];
            float outv;
            if      (epi == 1) outv = tanhf(a);
            else if (epi == 2) outv = a / (1.0f + __expf(-a));            // silu
            else if (epi == 3) outv = __expf(-__expf(bias[n] + a));       // decay
            else if (epi == 4) {                                          // x + xx*(tm+mix)
                long id = (long)m * ldc + n;
                outv = fmaf(e1[id], bias[n] + a, e0[id]);
            }
            else outv = a;
            Cc[(long)m * ldc + n] = outv;
        }
    }
}

// ----------------------------------------------------------------------------
// 3) WKV recurrence. One workgroup per (b,h): 256 threads = 4 k-quarters x 64
//    v-columns; 16 f32 state registers per thread. r/k/v/w staged via LDS.
// ----------------------------------------------------------------------------
__global__ __launch_bounds__(256)
void wkv_scan(const float* __restrict__ r, const float* __restrict__ k,
              const float* __restrict__ v, const float* __restrict__ w,
              const float* __restrict__ u, float* __restrict__ y)
{
    const int bh = blockIdx.x;
    const int b  = bh >> 4;          // / Hq
    const int h  = bh & 15;
    const int tid  = threadIdx.x;
    const int vcol = tid & 63;       // owned output column
    const int q    = tid >> 6;       // k-quarter 0..3

    __shared__ float rs[64], ks[64], wsh[64], vs[64], us[64], yp[4][64];

    float S[16];
    #pragma unroll
    for (int i = 0; i < 16; ++i) S[i] = 0.0f;
    if (tid < 64) us[tid] = u[h * HD + tid];
    __syncthreads();

    for (int t = 0; t < Tq; ++t) {
        const long base = ((long)b * Tq + t) * Cq + h * HD;
        if      (tid < 64)  rs [tid      ] = r[base + tid      ];
        else if (tid < 128) ks [tid - 64 ] = k[base + tid - 64 ];
        else if (tid < 192) vs [tid - 128] = v[base + tid - 128];
        else                wsh[tid - 192] = w[base + tid - 192];
        __syncthreads();

        const float vt = vs[vcol];
        float yacc = 0.0f;
        #pragma unroll
        for (int i = 0; i < 16; ++i) {
            const int kk = (q << 4) + i;
            const float at = ks[kk] * vt;
            const float s  = S[i];
            yacc = fmaf(rs[kk], fmaf(us[kk], at, s), yacc);   // y += r*(S + u*a)
            S[i] = fmaf(wsh[kk], s, at);                      // S = w*S + a
        }
        yp[q][vcol] = yacc;
        __syncthreads();
        if (q == 0)
            y[base + vcol] = yp[0][vcol] + yp[1][vcol] + yp[2][vcol] + yp[3][vcol];
        __syncthreads();
    }
}

// ----------------------------------------------------------------------------
// 4) Per-head GroupNorm (eps = 6.4e-4) * ln_w + ln_b, gated by silu(g).
// ----------------------------------------------------------------------------
__global__ __launch_bounds__(256)
void gn_gate(const float* __restrict__ y, const float* __restrict__ g,
             const float* __restrict__ lnw, const float* __restrict__ lnb,
             float* __restrict__ z2)
{
    int idx = blockIdx.x * blockDim.x + threadIdx.x;  // token*H + h
    if (idx >= BT * Hq) return;
    int h   = idx & (Hq - 1);
    int tok = idx >> 4;
    long base = (long)tok * Cq + h * HD;

    float s = 0.0f, ss = 0.0f;
    #pragma unroll 8
    for (int j = 0; j < HD; ++j) { float t = y[base + j]; s += t; ss = fmaf(t, t, ss); }
    float mu  = s * (1.0f / HD);
    float var = ss * (1.0f / HD) - mu * mu;
    float inv = rsqrtf(var + EPSGN);

    #pragma unroll 4
    for (int j = 0; j < HD; ++j) {
        int c = h * HD + j;
        float yn = fmaf((y[base + j] - mu) * inv, lnw[c], lnb[c]);
        z2[base + j] = yn * g[base + j];        // g already holds silu(xg @ Wg)
    }
}

// ----------------------------------------------------------------------------
// Host orchestration (graph-capture safe: launches only, all on `stream`).
// ----------------------------------------------------------------------------
extern "C" void kernel_launch(void* const* d_in, const int* in_sizes, int n_in,
                              void* d_out, int out_size, void* d_ws, size_t ws_size,
                              hipStream_t stream)
{
    const float* tmx    = (const float*)d_in[0];
    const float* tmw    = (const float*)d_in[1];
    const float* tmk    = (const float*)d_in[2];
    const float* tmv    = (const float*)d_in[3];
    const float* tmr    = (const float*)d_in[4];
    const float* tmg    = (const float*)d_in[5];
    const float* mw1    = (const float*)d_in[6];   // [C, 160]
    const float* mw2    = (const float*)d_in[7];   // [5, 32, C]
    const float* tdecay = (const float*)d_in[8];   // [C]
    const float* dw1    = (const float*)d_in[9];   // [C, 64]
    const float* dw2    = (const float*)d_in[10];  // [64, C]
    const float* u      = (const float*)d_in[11];  // [H, 64]
    const float* Wr     = (const float*)d_in[12];
    const float* Wk     = (const float*)d_in[13];
    const float* Wv     = (const float*)d_in[14];
    const float* Wg     = (const float*)d_in[15];
    const float* Wo     = (const float*)d_in[16];
    const float* lnw    = (const float*)d_in[17];
    const float* lnb    = (const float*)d_in[18];
    const float* x      = (const float*)d_in[19];  // [B, T, C]
    float* out = (float*)d_out;

    const long NTC = (long)BT * Cq;                // 4M floats
    float* ws   = (float*)d_ws;
    float* xx   = ws;                              // NTC
    float* zbuf = xx   + NTC;                      // NTC  (z, later reused for yn*g)
    float* xxx  = zbuf + NTC;                      // BT*160
    float* x5   = xxx  + (long)BT * (5 * DMIX);    // 5*NTC : xw,xk,xv,xr,xg
    float* rb   = x5   + 5 * NTC;                  // NTC
    float* kb   = rb   + NTC;
    float* vb   = kb   + NTC;
    float* gb   = vb   + NTC;
    float* wb   = gb   + NTC;
    float* dtmp = wb   + NTC;                      // BT*64

    const dim3 blk(256);

    // 1) token shift + maa_x mix
    prep_shift_mix<<<dim3((unsigned)((NTC + 255) / 256)), blk, 0, stream>>>(x, tmx, xx, zbuf);

    auto gemm = [&](const float* A, int lda, const float* Bm, int ldb,
                    float* Cc, int ldc, int M, int N, int K,
                    int epi, const float* bias, const float* e0, const float* e1) {
        dim3 grid((unsigned)((N + 63) / 64), (unsigned)((M + 127) / 128));
        wmma_gemm_f32<<<grid, blk, 0, stream>>>(A, lda, Bm, ldb, Cc, ldc,
                                                M, N, K, epi, bias, e0, e1);
    };

    // 2) xxx = tanh(z @ mw1)                               [4096 x 160]
    gemm(zbuf, Cq, mw1, 5 * DMIX, xxx, 5 * DMIX, BT, 5 * DMIX, Cq,
         1, nullptr, nullptr, nullptr);

    // 3) x? = x + xx*(tm? + xxx[:,l,:] @ mw2[l])  (mix fused in epilogue)
    const float* tms[5] = { tmw, tmk, tmv, tmr, tmg };
    for (int l = 0; l < 5; ++l)
        gemm(xxx + 32 * l, 5 * DMIX, mw2 + (long)l * DMIX * Cq, Cq,
             x5 + (long)l * NTC, Cq, BT, Cq, DMIX, 4, tms[l], x, xx);

    // 4) decay LoRA first (frees xw for reuse): dtmp = tanh(xw @ dw1)
    gemm(x5 + 0 * NTC, Cq, dw1, DDEC, dtmp, DDEC, BT, DDEC, Cq,
         1, nullptr, nullptr, nullptr);
    //    w = exp(-exp(tdecay + dtmp @ dw2))
    gemm(dtmp, DDEC, dw2, Cq, wb, Cq, BT, Cq, DDEC, 3, tdecay, nullptr, nullptr);

    // 5) projections r,k,v and gated g = silu(xg @ Wg)
    gemm(x5 + 3 * NTC, Cq, Wr, Cq, rb, Cq, BT, Cq, Cq, 0, nullptr, nullptr, nullptr);
    gemm(x5 + 1 * NTC, Cq, Wk, Cq, kb, Cq, BT, Cq, Cq, 0, nullptr, nullptr, nullptr);
    gemm(x5 + 2 * NTC, Cq, Wv, Cq, vb, Cq, BT, Cq, Cq, 0, nullptr, nullptr, nullptr);
    gemm(x5 + 4 * NTC, Cq, Wg, Cq, gb, Cq, BT, Cq, Cq, 2, nullptr, nullptr, nullptr);

    // 6) sequential WKV scan  (y reuses the now-dead xw slab)
    float* yb = x5;
    wkv_scan<<<dim3(Bq * Hq), dim3(256), 0, stream>>>(rb, kb, vb, wb, u, yb);

    // 7) GroupNorm + gate -> zbuf (reused)
    gn_gate<<<dim3((BT * Hq + 255) / 256), blk, 0, stream>>>(yb, gb, lnw, lnb, zbuf);

    // 8) out = (yn * g) @ Wo
    gemm(zbuf, Cq, Wo, Cq, out, Cq, BT, Cq, Cq, 0, nullptr, nullptr, nullptr);

    (void)in_sizes; (void)n_in; (void)out_size; (void)ws_size;
}